// GNNEncoder_8581344657809
// MI455X (gfx1250) — compile-verified
//
#include <hip/hip_runtime.h>
#include <hip/hip_bf16.h>

typedef __attribute__((ext_vector_type(2))) float v2f;
typedef __attribute__((ext_vector_type(8))) float v8f;

#define HD 128          // feature width (D == H == 128)
#define EPS 1e-5f

// ---------------------------------------------------------------- utility
__global__ void fill_zero_kernel(float* __restrict__ p, long n) {
    long t = (long)blockIdx.x * blockDim.x + threadIdx.x;
    if (t < n) p[t] = 0.0f;
}

__global__ void degree_kernel(const int* __restrict__ idx, float* __restrict__ deg, int E) {
    int t = blockIdx.x * blockDim.x + threadIdx.x;
    if (t < E) unsafeAtomicAdd(&deg[idx[t]], 1.0f);
}

__global__ void rsqrt_kernel(float* __restrict__ d, long n) {
    long t = (long)blockIdx.x * blockDim.x + threadIdx.x;
    if (t < n) d[t] = rsqrtf(fmaxf(d[t], 1.0f));
}

// per-edge combined norm: rsqrt(deg_out[src]) * rsqrt(deg_in[dst])
__global__ void escale_kernel(const int* __restrict__ src, const int* __restrict__ dst,
                              const float* __restrict__ rout, const float* __restrict__ rin,
                              float* __restrict__ es, int E) {
    int t = blockIdx.x * blockDim.x + threadIdx.x;
    if (t < E) es[t] = rout[src[t]] * rin[dst[t]];
}

__global__ void bias_sum_kernel(const float* __restrict__ bg, float* __restrict__ bsum) {
    int c = threadIdx.x;
    if (c < HD) bsum[c] = bg[c] + bg[HD + c] + bg[2 * HD + c];
}

// ------------------------------------------------------- edge scatter-add
// one wave per edge; each lane moves 4 contiguous floats (float4 gather,
// 4 hardware f32 atomic adds). agg row stride = 384 (3 relations concat).
__global__ void scatter_kernel(const float* __restrict__ h, const int* __restrict__ src,
                               const int* __restrict__ dst, const float* __restrict__ escale,
                               float* __restrict__ agg, int E, int colBase) {
    int e    = blockIdx.x * (blockDim.x >> 5) + (threadIdx.x >> 5);
    int lane = threadIdx.x & 31;
    if (e >= E) return;
    int   s  = src[e];
    int   d  = dst[e];
    float sc = escale[e];
    const float4 hv = *(const float4*)(h + (size_t)s * HD + lane * 4);
    float* ap = agg + (size_t)d * (3 * HD) + colBase + lane * 4;
    unsafeAtomicAdd(ap + 0, hv.x * sc);
    unsafeAtomicAdd(ap + 1, hv.y * sc);
    unsafeAtomicAdd(ap + 2, hv.z * sc);
    unsafeAtomicAdd(ap + 3, hv.w * sc);
}

// ------------------------------------------------------------ WMMA GEMM
// out[n,128] = epilogue(A[n,K] @ W[K,128] + bias)
//   mode 0: + bias
//   mode 1: relu then eval-BatchNorm
// 256 threads = 8 waves; wave w -> rows [blk*128 + w*16, +16), all 128 cols.
// W staged through 64KB LDS in K-chunks of 128, stored K-PAIR INTERLEAVED:
//   ldsW[(k>>1)*256 + col*2 + (k&1)] = W[k][col]
// so each B fragment (W[k'][col], W[k'+1][col]) is one aligned ds_load_b64
// into an even VGPR pair -> no repack movs before v_wmma.
__global__ void gemm_wmma_kernel(const float* __restrict__ A, const float* __restrict__ W,
                                 const float* __restrict__ bias, float* __restrict__ out,
                                 int n, int K, int mode,
                                 const float* __restrict__ gamma, const float* __restrict__ beta,
                                 const float* __restrict__ mean,  const float* __restrict__ var) {
    __shared__ float ldsW[HD * HD];   // 64 KB

    const int tid  = threadIdx.x;
    const int wave = tid >> 5;
    const int lane = tid & 31;
    const int m    = lane & 15;       // M (A-frag) / N (B-frag) index
    const int half = lane >> 4;       // K sub-slot selector

    const int row0 = blockIdx.x * 128 + wave * 16;
    int aRow = row0 + m;
    if (aRow >= n) aRow = n - 1;      // clamp: keep wave converged for WMMA
    const float* Arow = A + (size_t)aRow * K;

    // lane-invariant-per-k base into the pair-interleaved LDS image:
    // B frag for (k, c): ldsB + (k>>1)*256 + c*32   (dwords), 8B aligned
    const float* ldsB = ldsW + (half << 8) + (m << 1);

    v8f acc[8] = {};                  // 8 tiles of 16x16 -> 16 rows x 128 cols

    for (int kb = 0; kb < K; kb += HD) {
        if (kb + HD < K) __builtin_prefetch(Arow + kb + HD, 0, 3);  // next A chunk

        // stage W[kb..kb+127][0..127] into LDS, K-pair interleaved
        for (int i = tid; i < HD * HD; i += blockDim.x) {
            int kr  = i >> 7;
            int col = i & 127;
            ldsW[((kr >> 1) << 8) + (col << 1) + (kr & 1)] = W[(size_t)kb * HD + i];
        }
        __syncthreads();

        for (int k = 0; k < HD; k += 4) {
            // A frag 16x4: lanes 0-15 hold K=k,k+1 ; lanes 16-31 hold K=k+2,k+3
            v2f a = *(const v2f*)(Arow + kb + k + 2 * half);
            const float* bp = ldsB + ((k >> 1) << 8);
#pragma unroll
            for (int c = 0; c < 8; ++c) {
                v2f b = *(const v2f*)(bp + c * 32);
                acc[c] = __builtin_amdgcn_wmma_f32_16x16x4_f32(
                    false, a, false, b, (short)0, acc[c], false, false);
            }
        }
        __syncthreads();
    }

    // epilogue: D element (row0 + v + 8*half, c*16 + m) = acc[c][v]
#pragma unroll
    for (int c = 0; c < 8; ++c) {
        const int col  = c * 16 + m;
        const float bv = bias[col];
        float g = 0.f, bt = 0.f, mu = 0.f, iv = 0.f;
        if (mode == 1) {
            g  = gamma[col];
            bt = beta[col];
            mu = mean[col];
            iv = rsqrtf(var[col] + EPS);
        }
#pragma unroll
        for (int v = 0; v < 8; ++v) {
            int row = row0 + v + 8 * half;
            if (row < n) {
                float val = acc[c][v] + bv;
                if (mode == 1) {
                    val = fmaxf(val, 0.0f);
                    val = (val - mu) * iv * g + bt;
                }
                out[(size_t)row * HD + col] = val;
            }
        }
    }
}

// ---------------------------------------------------------------- driver
extern "C" void kernel_launch(void* const* d_in, const int* in_sizes, int n_in,
                              void* d_out, int out_size, void* d_ws, size_t ws_size,
                              hipStream_t stream) {
    const float* x = (const float*)d_in[0];
    const int* srcs[3] = {(const int*)d_in[1], (const int*)d_in[3], (const int*)d_in[5]};
    const int* dsts[3] = {(const int*)d_in[2], (const int*)d_in[4], (const int*)d_in[6]};
    const float* Wg[2] = {(const float*)d_in[7], (const float*)d_in[9]};   // [3,128,128] == [384,128]
    const float* bg[2] = {(const float*)d_in[8], (const float*)d_in[10]};  // [3,128]
    const float* Wfc   = (const float*)d_in[11];                           // [2,128,128]
    const float* bfc   = (const float*)d_in[12];                           // [2,128]
    const float* gamma = (const float*)d_in[13];
    const float* beta  = (const float*)d_in[14];
    const float* mean  = (const float*)d_in[15];
    const float* var   = (const float*)d_in[16];
    float* out = (float*)d_out;

    const int N = in_sizes[0] / HD;   // 100000
    const int E = in_sizes[1];        // 1600000

    // workspace carve-up (floats)
    float* ws = (float*)d_ws;
    size_t off = 0;
    float* deg  = ws + off; off += (size_t)6 * N;        // [rout0..2 | rin0..2]
    float* es   = ws + off; off += (size_t)3 * E;        // per-edge combined scale
    float* agg  = ws + off; off += (size_t)N * 3 * HD;   // [N,384] concat relations
    float* z    = ws + off; off += (size_t)N * HD;       // gconv-sum output
    float* h1   = ws + off; off += (size_t)N * HD;       // layer-0 hidden
    float* bsum = ws + off; off += HD;                   // summed gconv bias

    auto cdiv = [](long a, long b) { return (int)((a + b - 1) / b); };

    // ---- degrees (static across layers) -> rsqrt -> per-edge scale
    fill_zero_kernel<<<cdiv((long)6 * N, 256), 256, 0, stream>>>(deg, (long)6 * N);
    for (int r = 0; r < 3; ++r) {
        degree_kernel<<<cdiv(E, 256), 256, 0, stream>>>(srcs[r], deg + (size_t)r * N, E);
        degree_kernel<<<cdiv(E, 256), 256, 0, stream>>>(dsts[r], deg + (size_t)(3 + r) * N, E);
    }
    rsqrt_kernel<<<cdiv((long)6 * N, 256), 256, 0, stream>>>(deg, (long)6 * N);
    for (int r = 0; r < 3; ++r) {
        escale_kernel<<<cdiv(E, 256), 256, 0, stream>>>(
            srcs[r], dsts[r], deg + (size_t)r * N, deg + (size_t)(3 + r) * N,
            es + (size_t)r * E, E);
    }

    // ---- two GNN layers
    const float* h = x;
    const int gemmGrid = cdiv(N, 128);
    for (int L = 0; L < 2; ++L) {
        fill_zero_kernel<<<cdiv((long)N * 3 * HD, 256), 256, 0, stream>>>(agg, (long)N * 3 * HD);
        for (int r = 0; r < 3; ++r) {
            scatter_kernel<<<cdiv(E, 8), 256, 0, stream>>>(
                h, srcs[r], dsts[r], es + (size_t)r * E, agg, E, r * HD);
        }
        bias_sum_kernel<<<1, 128, 0, stream>>>(bg[L], bsum);

        // z = agg[N,384] @ Wg_flat[384,128] + (b0+b1+b2)
        gemm_wmma_kernel<<<gemmGrid, 256, 0, stream>>>(
            agg, Wg[L], bsum, z, N, 3 * HD, 0,
            nullptr, nullptr, nullptr, nullptr);

        // h' = BN(relu(z @ Wfc[L] + bfc[L]))
        float* hn = (L == 0) ? h1 : out;
        gemm_wmma_kernel<<<gemmGrid, 256, 0, stream>>>(
            z, Wfc + (size_t)L * HD * HD, bfc + (size_t)L * HD, hn, N, HD, 1,
            gamma + (size_t)L * HD, beta + (size_t)L * HD,
            mean + (size_t)L * HD, var + (size_t)L * HD);
        h = hn;
    }
}